// ScatterImpl_90314572300821
// MI455X (gfx1250) — compile-verified
//
#include <hip/hip_runtime.h>
#include <hip/hip_bf16.h>

// Scatter-sum: out[index[e], :] += src[e, :]
//   E = in_sizes[1] edges, D = 64 features (f32), N = out_size / 64 nodes.
// Strategy: build CSR (counts -> scan -> fill) with cheap integer atomics,
// then an atomic-free per-node gather-reduce using 128-bit loads with
// cooperative edge-id fetch (breaks the pointer-chase dependence chain).

#define DFEAT 64            // floats per row
#define DQ    16            // float4 per row (64/4); also lanes per node
#define SCAN_T 1024

// ---------------- Phase A: zero counts ----------------
__global__ void zero_counts_kernel(int* __restrict__ cnt, int n) {
    int i = blockIdx.x * blockDim.x + threadIdx.x;
    if (i < n) cnt[i] = 0;
}

// ---------------- Phase B: histogram (integer atomics) ----------------
__global__ void hist_kernel(const int* __restrict__ index,
                            int* __restrict__ cnt, int n_edges) {
    int e = blockIdx.x * blockDim.x + threadIdx.x;
    if (e < n_edges) {
        atomicAdd(&cnt[index[e]], 1);
    }
}

// ---------------- Phase C: single-block exclusive scan over counts ----
// Writes offs[i] = exclusive prefix sum, cursor[i] = same (consumed in fill).
__global__ void scan_kernel(const int* __restrict__ cnt,
                            int* __restrict__ offs,
                            int* __restrict__ cursor, int n_nodes) {
    __shared__ int sdata[SCAN_T];
    const int tid   = threadIdx.x;
    const int chunk = (n_nodes + SCAN_T - 1) / SCAN_T;
    const int begin = tid * chunk;
    const int end   = (begin + chunk < n_nodes) ? (begin + chunk) : n_nodes;

    // local chunk sum
    int sum = 0;
    for (int i = begin; i < end; ++i) sum += cnt[i];
    sdata[tid] = sum;
    __syncthreads();

    // Hillis-Steele inclusive scan over thread sums (in LDS)
    for (int d = 1; d < SCAN_T; d <<= 1) {
        int v = (tid >= d) ? sdata[tid - d] : 0;
        __syncthreads();
        sdata[tid] += v;
        __syncthreads();
    }

    // exclusive base for this thread's chunk
    int run = sdata[tid] - sum;
    for (int i = begin; i < end; ++i) {
        offs[i]   = run;
        cursor[i] = run;
        run += cnt[i];
    }
}

// ---------------- Phase D: fill CSR edge lists (integer atomics) -------
__global__ void fill_kernel(const int* __restrict__ index,
                            int* __restrict__ cursor,
                            int* __restrict__ edge_ids, int n_edges) {
    int e = blockIdx.x * blockDim.x + threadIdx.x;
    if (e < n_edges) {
        int pos = atomicAdd(&cursor[index[e]], 1);  // absolute position
        edge_ids[pos] = e;
    }
}

// ---------------- Phase E: atomic-free gather-reduce -------------------
// 16 lanes own one node: lane handles 4 contiguous floats (float4 ->
// global_load_b128; the 16 lanes together cover one 256B row).
// Edge ids are fetched cooperatively: 16 ids per coalesced access, then
// broadcast with __shfl(width=16). Inner loop has no load->load dependence,
// so up to 16 b128 row loads are in flight per group. Next chunk's ids are
// software-pipelined one chunk ahead. Output written exactly once.
__global__ void reduce_kernel(const float* __restrict__ src,
                              const int* __restrict__ offs,
                              const int* __restrict__ cnt,
                              const int* __restrict__ edge_ids,
                              float* __restrict__ out, int n_nodes) {
    const int lane = threadIdx.x & (DQ - 1);
    const int node = (blockIdx.x * blockDim.x + threadIdx.x) >> 4;
    if (node >= n_nodes) return;

    const int start = offs[node];
    const int count = cnt[node];
    const float4* __restrict__ s4 = (const float4*)src;

    float4 acc = make_float4(0.f, 0.f, 0.f, 0.f);

    // Preload first chunk of up to 16 edge ids (one 4B load/lane, coalesced).
    int eid = (lane < count) ? edge_ids[start + lane] : 0;

    for (int j0 = 0; j0 < count; j0 += DQ) {
        // Software pipeline: fetch next chunk's ids while consuming this one.
        const int pos      = j0 + DQ + lane;
        const int eid_next = (pos < count) ? edge_ids[start + pos] : 0;

        // gfx1250 global_prefetch_b8: warm L2 for next chunk's gathered rows.
        __builtin_prefetch(&s4[(size_t)eid_next * DQ + lane], 0, 3);

        const int nrem = (count - j0 < DQ) ? (count - j0) : DQ;
        for (int j = 0; j < nrem; ++j) {
            const int e = __shfl(eid, j, DQ);       // broadcast within 16-lane group
            const float4 v = s4[(size_t)e * DQ + lane];  // global_load_b128
            acc.x += v.x; acc.y += v.y; acc.z += v.z; acc.w += v.w;
        }
        eid = eid_next;
    }
    ((float4*)out)[(size_t)node * DQ + lane] = acc;  // global_store_b128
}

extern "C" void kernel_launch(void* const* d_in, const int* in_sizes, int n_in,
                              void* d_out, int out_size, void* d_ws, size_t ws_size,
                              hipStream_t stream) {
    const float* src   = (const float*)d_in[0];
    const int*   index = (const int*)d_in[1];   // harness: integer -> const int*
    float*       out   = (float*)d_out;

    const int n_edges = in_sizes[1];            // 800000
    const int n_nodes = out_size / DFEAT;       // 50000

    // Workspace layout (all int32): counts | offsets | cursor | edge_ids
    int* cnt      = (int*)d_ws;
    int* offs     = cnt    + n_nodes;
    int* cursor   = offs   + n_nodes;
    int* edge_ids = cursor + n_nodes;
    (void)ws_size; (void)n_in;

    const int TB = 256;

    // A: zero counters
    zero_counts_kernel<<<(n_nodes + TB - 1) / TB, TB, 0, stream>>>(cnt, n_nodes);

    // B: histogram
    hist_kernel<<<(n_edges + TB - 1) / TB, TB, 0, stream>>>(index, cnt, n_edges);

    // C: scan counts -> offsets (+ cursor copy); single block
    scan_kernel<<<1, SCAN_T, 0, stream>>>(cnt, offs, cursor, n_nodes);

    // D: fill per-node edge lists
    fill_kernel<<<(n_edges + TB - 1) / TB, TB, 0, stream>>>(index, cursor, edge_ids, n_edges);

    // E: gather-reduce, 16 lanes per node
    const long threads_needed = (long)n_nodes * DQ;
    reduce_kernel<<<(int)((threads_needed + TB - 1) / TB), TB, 0, stream>>>(
        src, offs, cnt, edge_ids, out, n_nodes);
}